// SparseProp_67284957659449
// MI455X (gfx1250) — compile-verified
//
#include <hip/hip_runtime.h>
#include <hip/hip_bf16.h>
#include <math.h>

// ---------------------------------------------------------------------------
// GCN propagation: out = D^-1/2 (A + A^T) D^-1/2 x     (N=50000, E=800000, D=64)
//
// CSR built on device; one wave32 per output row. Per 4-neighbor chunk the
// multiply-accumulate runs through V_WMMA_F32_16X16X4_F32: A(16x4) holds the 4
// edge coefficients replicated across all 16 M-rows, B(4x16) holds 16 features
// of the 4 gathered x rows; every D row is the same sum, row M=0 is read out.
// Row bounds are forced into SGPRs (readfirstlane) so the chunk loop is a
// uniform scalar loop and EXEC is provably all-ones at every WMMA.
// ---------------------------------------------------------------------------

typedef float v2f __attribute__((ext_vector_type(2)));
typedef float v8f __attribute__((ext_vector_type(8)));

#define DFEAT 64

__global__ void gcn_zero_i32(int* p, int n) {
    int i = blockIdx.x * blockDim.x + threadIdx.x;
    if (i < n) p[i] = 0;
}

__global__ void gcn_zero_f32(float* p, int n) {
    int i = blockIdx.x * blockDim.x + threadIdx.x;
    if (i < n) p[i] = 0.0f;
}

// deg[i] = #occurrences of i in src of symmetrized edge list
__global__ void gcn_count_deg(const int* __restrict__ eu, const int* __restrict__ ev,
                              int* __restrict__ deg, int E) {
    int e = blockIdx.x * blockDim.x + threadIdx.x;
    if (e < E) {
        atomicAdd(&deg[eu[e]], 1);
        atomicAdd(&deg[ev[e]], 1);
    }
}

// Single-block exclusive scan over deg -> row_ptr[N+1]; also emits per-row
// insert cursors and inv_sqrt(max(deg,1)). 50000 elems -> 49 chunk iterations.
__global__ void gcn_scan(const int* __restrict__ deg, int* __restrict__ row_ptr,
                         int* __restrict__ cursor, float* __restrict__ isq, int n) {
    __shared__ int sdata[1024];
    __shared__ int carry;
    if (threadIdx.x == 0) carry = 0;
    __syncthreads();
    for (int base = 0; base < n; base += 1024) {
        int i = base + (int)threadIdx.x;
        int v = (i < n) ? deg[i] : 0;
        sdata[threadIdx.x] = v;
        __syncthreads();
        for (int off = 1; off < 1024; off <<= 1) {   // Hillis-Steele inclusive scan
            int t = (threadIdx.x >= (unsigned)off) ? sdata[threadIdx.x - off] : 0;
            __syncthreads();
            sdata[threadIdx.x] += t;
            __syncthreads();
        }
        int incl = sdata[threadIdx.x];
        int excl = incl - v;
        if (i < n) {
            int rp = carry + excl;
            row_ptr[i] = rp;
            cursor[i]  = rp;
            float d    = (float)(v < 1 ? 1 : v);
            isq[i]     = 1.0f / sqrtf(d);
        }
        __syncthreads();
        if (threadIdx.x == 1023) carry += incl;
        __syncthreads();
    }
    if (threadIdx.x == 0) row_ptr[n] = carry;
}

__global__ void gcn_fill_csr(const int* __restrict__ eu, const int* __restrict__ ev,
                             int* __restrict__ cursor, int* __restrict__ colv, int E) {
    int e = blockIdx.x * blockDim.x + threadIdx.x;
    if (e < E) {
        int u = eu[e], v = ev[e];
        int p = atomicAdd(&cursor[u], 1);
        colv[p] = v;
        int q = atomicAdd(&cursor[v], 1);
        colv[q] = u;
    }
}

// One wave32 per destination row. 4 feature tiles of 16 accumulated in v8f.
// K lane layout (ISA 7.12.2): lanes 0-15 hold K={0,1} in VGPR{0,1};
// lanes 16-31 hold K={2,3}. Tail neighbors are handled by clamping the gather
// index to a valid element and zeroing the coefficient (no divergent loads).
__global__ void gcn_spmm_wmma(const float* __restrict__ x,
                              const int* __restrict__ row_ptr,
                              const int* __restrict__ colv,
                              const float* __restrict__ isq,
                              float* __restrict__ out, int n) {
    int wave = (int)((blockIdx.x * blockDim.x + threadIdx.x) >> 5);
    int lane = (int)(threadIdx.x & 31u);
    if (wave >= n) return;                 // wave-uniform: whole wave exits together
    const int r = wave;

    // Row bounds / self scale are wave-uniform: pin them to SGPRs so the chunk
    // loop is a scalar s_cmp/s_cbranch loop (EXEC stays all-ones for WMMA).
    const int   beg = __builtin_amdgcn_readfirstlane(row_ptr[r]);
    const int   end = __builtin_amdgcn_readfirstlane(row_ptr[r + 1]);
    const float ir  = __int_as_float(__builtin_amdgcn_readfirstlane(__float_as_int(isq[r])));

    const int lhalf = lane >> 4;           // 0: K-pair {0,1}   1: K-pair {2,3}
    const int nlo   = lane & 15;           // feature index within a 16-tile

    v8f acc0 = {}, acc1 = {}, acc2 = {}, acc3 = {};

    for (int j = beg; j < end; j += 4) {
        int j0 = j + 2 * lhalf;
        int j1 = j0 + 1;
        // clamp to a valid element (end > beg inside the loop), select coeff=0
        int jc0 = (j0 < end) ? j0 : beg;
        int jc1 = (j1 < end) ? j1 : beg;
        int c0 = colv[jc0];
        int c1 = colv[jc1];
        float s0 = isq[c0];
        float s1 = isq[c1];
        float a0 = (j0 < end) ? ir * s0 : 0.0f;
        float a1 = (j1 < end) ? ir * s1 : 0.0f;
        v2f A; A.x = a0; A.y = a1;

        const float* xr0 = x + (size_t)c0 * DFEAT;
        const float* xr1 = x + (size_t)c1 * DFEAT;
        v2f B0, B1, B2, B3;                 // four 16-feature column tiles
        B0.x = xr0[nlo];        B0.y = xr1[nlo];
        B1.x = xr0[16 + nlo];   B1.y = xr1[16 + nlo];
        B2.x = xr0[32 + nlo];   B2.y = xr1[32 + nlo];
        B3.x = xr0[48 + nlo];   B3.y = xr1[48 + nlo];

        acc0 = __builtin_amdgcn_wmma_f32_16x16x4_f32(false, A, false, B0, (short)0, acc0, false, false);
        acc1 = __builtin_amdgcn_wmma_f32_16x16x4_f32(false, A, false, B1, (short)0, acc1, false, false);
        acc2 = __builtin_amdgcn_wmma_f32_16x16x4_f32(false, A, false, B2, (short)0, acc2, false, false);
        acc3 = __builtin_amdgcn_wmma_f32_16x16x4_f32(false, A, false, B3, (short)0, acc3, false, false);
    }

    if (lhalf == 0) {                       // divergence only after all WMMAs
        float* o = out + (size_t)r * DFEAT + nlo;
        o[0]  = acc0[0];
        o[16] = acc1[0];
        o[32] = acc2[0];
        o[48] = acc3[0];
    }
}

// -------- fallback path (tiny workspace): atomic scatter, wave per edge -----
__global__ void gcn_inv_sqrt(const int* __restrict__ deg, float* __restrict__ isq, int n) {
    int i = blockIdx.x * blockDim.x + threadIdx.x;
    if (i < n) {
        int d = deg[i];
        isq[i] = 1.0f / sqrtf((float)(d < 1 ? 1 : d));
    }
}

__global__ void gcn_scatter(const float* __restrict__ x, const int* __restrict__ eu,
                            const int* __restrict__ ev, const float* __restrict__ isq,
                            float* __restrict__ out, int E) {
    int wave = (int)((blockIdx.x * blockDim.x + threadIdx.x) >> 5);
    int lane = (int)(threadIdx.x & 31u);
    if (wave >= E) return;
    int u = eu[wave], v = ev[wave];
    float val = isq[u] * isq[v];
    const float* xu = x + (size_t)u * DFEAT;
    const float* xv = x + (size_t)v * DFEAT;
    float* ou = out + (size_t)u * DFEAT;
    float* ov = out + (size_t)v * DFEAT;
    atomicAdd(&ou[lane],      val * xv[lane]);
    atomicAdd(&ou[lane + 32], val * xv[lane + 32]);
    atomicAdd(&ov[lane],      val * xu[lane]);
    atomicAdd(&ov[lane + 32], val * xu[lane + 32]);
}

extern "C" void kernel_launch(void* const* d_in, const int* in_sizes, int n_in,
                              void* d_out, int out_size, void* d_ws, size_t ws_size,
                              hipStream_t stream) {
    const float* x  = (const float*)d_in[0];
    const int*   ei = (const int*)d_in[1];      // [2, E] row-major int
    const int N = in_sizes[0] / DFEAT;          // 50000
    const int E = in_sizes[1] / 2;              // 800000
    const int* eu = ei;                          // row 0
    const int* ev = ei + E;                      // row 1
    float* out = (float*)d_out;

    // workspace layout
    int*   deg    = (int*)d_ws;                  // N
    int*   cursor = deg + N;                     // N
    float* isq    = (float*)(cursor + N);        // N
    int*   rowp   = (int*)(isq + N);             // N+1
    int*   colv   = rowp + (N + 1);              // 2E
    size_t need   = (size_t)(3 * N + (N + 1) + 2 * E) * sizeof(int);

    const int TB = 256;
    gcn_zero_i32<<<(N + TB - 1) / TB, TB, 0, stream>>>(deg, N);
    gcn_count_deg<<<(E + TB - 1) / TB, TB, 0, stream>>>(eu, ev, deg, E);

    if (ws_size >= need) {
        gcn_scan<<<1, 1024, 0, stream>>>(deg, rowp, cursor, isq, N);
        gcn_fill_csr<<<(E + TB - 1) / TB, TB, 0, stream>>>(eu, ev, cursor, colv, E);
        long long threads = (long long)N * 32;   // one wave32 per row
        int blocks = (int)((threads + TB - 1) / TB);
        gcn_spmm_wmma<<<blocks, TB, 0, stream>>>(x, rowp, colv, isq, out, N);
    } else {
        // small-workspace fallback: atomic scatter (deg + isq only)
        float* isq2 = (float*)(deg + N);
        gcn_inv_sqrt<<<(N + TB - 1) / TB, TB, 0, stream>>>(deg, isq2, N);
        gcn_zero_f32<<<(N * DFEAT + TB - 1) / TB, TB, 0, stream>>>(out, N * DFEAT);
        long long threads = (long long)E * 32;
        int blocks = (int)((threads + TB - 1) / TB);
        gcn_scatter<<<blocks, TB, 0, stream>>>(x, eu, ev, isq2, out, E);
    }
}